// GraphConvEncoder_45818711113979
// MI455X (gfx1250) — compile-verified
//
#include <hip/hip_runtime.h>
#include <hip/hip_bf16.h>
#include <math.h>

// ---------------- problem constants (match reference) ----------------
#define NN      524288      // total nodes (B*NPG)
#define BB      1024        // graphs
#define NPG     512         // nodes per graph
#define DD      8           // feature dim
#define GATE    64          // attention gate dim
#define NEGF    (-1e30f)

typedef __attribute__((ext_vector_type(2))) float v2f;
typedef __attribute__((ext_vector_type(8))) float v8f;

static __device__ __forceinline__ v8f wmma_f32_16x16x4(v2f a, v2f b, v8f c) {
  // D = A(16x4,f32) * B(4x16,f32) + C(16x16,f32)   [VOP3P, wave32]
  return __builtin_amdgcn_wmma_f32_16x16x4_f32(false, a, false, b, (short)0, c,
                                               false, false);
}

// ---------------- small elementwise kernels ----------------
__global__ void init_state_kernel(const float* __restrict__ x,
                                  float* __restrict__ xcur,
                                  float* __restrict__ maskf,
                                  float* __restrict__ out) {
  int i = blockIdx.x * blockDim.x + threadIdx.x;        // over N*8
  xcur[i] = x[i];
  if (i < NN) maskf[i] = 1.0f;
  if (i < BB * GATE) out[i] = 0.0f;
}

__global__ void deg_init_kernel(const float* __restrict__ maskf,
                                float* __restrict__ deg) {
  int n = blockIdx.x * blockDim.x + threadIdx.x;
  deg[n] = maskf[n];                                    // self-loop contribution
}

__global__ void edge_deg_kernel(const int* __restrict__ row,
                                const int* __restrict__ col,
                                const float* __restrict__ maskf,
                                float* __restrict__ deg, int E) {
  int e = blockIdx.x * blockDim.x + threadIdx.x;
  if (e >= E) return;
  int r = row[e], c = col[e];
  float w = maskf[r] * maskf[c];
  if (w != 0.0f) atomicAdd(&deg[c], w);
}

__global__ void dinv_kernel(float* __restrict__ deg) {
  int n = blockIdx.x * blockDim.x + threadIdx.x;
  float d = deg[n];
  // unmasked nodes have deg==0 -> dinv==0, which implicitly carries the mask
  deg[n] = (d > 0.0f) ? __frsqrt_rn(d) : 0.0f;          // deg buffer now holds dinv
}

__global__ void agg_init_kernel(const float* __restrict__ ht,
                                const float* __restrict__ dinv,
                                float* __restrict__ agg) {
  int i = blockIdx.x * blockDim.x + threadIdx.x;        // over N*8
  int n = i >> 3;
  float di = dinv[n];                                   // 0 for unmasked nodes
  agg[i] = ht[i] * (di * di);                           // self-loop term
}

__global__ void edge_scatter_kernel(const int* __restrict__ row,
                                    const int* __restrict__ col,
                                    const float* __restrict__ dinv,
                                    const float* __restrict__ ht,
                                    float* __restrict__ agg, int E) {
  int e = blockIdx.x * blockDim.x + threadIdx.x;
  if (e >= E) return;
  int r = row[e], c = col[e];
  // dinv==0 for unmasked endpoints, so norm==0 exactly when the edge is pruned
  float norm = dinv[r] * dinv[c];
  if (norm == 0.0f) return;
  const float4* h4 = (const float4*)(ht + (size_t)r * DD);
  float4 h0 = h4[0], h1 = h4[1];
  float* ag = agg + (size_t)c * DD;
  atomicAdd(ag + 0, h0.x * norm);
  atomicAdd(ag + 1, h0.y * norm);
  atomicAdd(ag + 2, h0.z * norm);
  atomicAdd(ag + 3, h0.w * norm);
  atomicAdd(ag + 4, h1.x * norm);
  atomicAdd(ag + 5, h1.y * norm);
  atomicAdd(ag + 6, h1.z * norm);
  atomicAdd(ag + 7, h1.w * norm);
}

__global__ void bias_relu_kernel(const float* __restrict__ agg,
                                 const float* __restrict__ b,
                                 const float* __restrict__ maskf,
                                 float* __restrict__ xcur) {
  int i = blockIdx.x * blockDim.x + threadIdx.x;        // over N*8
  int n = i >> 3, d = i & 7;
  xcur[i] = maskf[n] * fmaxf(0.0f, agg[i] + b[d]);
}

__global__ void score_kernel(const float* __restrict__ xcur,
                             const float* __restrict__ p,
                             float* __restrict__ score) {
  int n = blockIdx.x * blockDim.x + threadIdx.x;
  float pn2 = 0.0f, dot = 0.0f;
#pragma unroll
  for (int d = 0; d < DD; ++d) {
    float pv = p[d];                                    // uniform -> SMEM loads
    pn2 += pv * pv;
    dot += xcur[(size_t)n * DD + d] * pv;
  }
  score[n] = tanhf(dot * __frsqrt_rn(pn2));
}

// ---------------- TopK pooling: exact rank by counting ----------------
__global__ __launch_bounds__(NPG) void topk_kernel(const float* __restrict__ score,
                                                   float* __restrict__ maskf,
                                                   float* __restrict__ xcur,
                                                   int k) {
  __shared__ float ss[NPG];
  int g = blockIdx.x, t = threadIdx.x;
  int n = g * NPG + t;
  float mk = maskf[n];
  float sraw = score[n];
  float sm = (mk > 0.0f) ? sraw : NEGF;
  ss[t] = sm;
  __syncthreads();
  int rank = 0;
#pragma unroll 8
  for (int j = 0; j < NPG; ++j) {
    float sj = ss[j];
    rank += (sj > sm) || (sj == sm && j < t);           // stable desc order
  }
  bool keep = (mk > 0.0f) && (rank < k);
  maskf[n] = keep ? 1.0f : 0.0f;
  float f = keep ? sraw : 0.0f;
  float4* xr = (float4*)(xcur + (size_t)n * DD);
  float4 a = xr[0], b = xr[1];
  a.x *= f; a.y *= f; a.z *= f; a.w *= f;
  b.x *= f; b.y *= f; b.z *= f; b.w *= f;
  xr[0] = a; xr[1] = b;
}

// ---------------- dense GEMM h = x @ W (8x8) via WMMA ----------------
__global__ __launch_bounds__(256) void gemm_xW_kernel(const float* __restrict__ x,
                                                      const float* __restrict__ W,
                                                      float* __restrict__ ht) {
  int gwave = (blockIdx.x * 256 + threadIdx.x) >> 5;    // one 16-node tile / wave
  int lane  = threadIdx.x & 31;
  int m0    = gwave * 16;
  int lhi   = lane >> 4;                                // 0 | 1 half-wave
  int khalf = lhi * 2;
  int m     = m0 + (lane & 15);
  int c     = lane & 15;

  // A 16x4 f32 layout: lanes0-15 -> K0,K1 ; lanes16-31 -> K2,K3
  v2f a0, a1;
  a0.x = x[(size_t)m * DD + khalf + 0];
  a0.y = x[(size_t)m * DD + khalf + 1];
  a1.x = x[(size_t)m * DD + 4 + khalf + 0];
  a1.y = x[(size_t)m * DD + 4 + khalf + 1];
  // B 4x16 f32 layout: vgpr r, lanes0-15 -> K=r ; lanes16-31 -> K=r+2.
  // Cols 8..15 zero-padded via select (clamped unconditional loads keep EXEC
  // untouched — no saveexec divergence ahead of the WMMAs).
  int cc = c & (DD - 1);
  float zsel = (c < DD) ? 1.0f : 0.0f;
  v2f b0, b1;
  b0.x = W[(khalf + 0) * DD + cc] * zsel;
  b0.y = W[(khalf + 1) * DD + cc] * zsel;
  b1.x = W[(4 + khalf + 0) * DD + cc] * zsel;
  b1.y = W[(4 + khalf + 1) * DD + cc] * zsel;

  v8f d = {};
  d = wmma_f32_16x16x4(a0, b0, d);
  d = wmma_f32_16x16x4(a1, b1, d);

  if (c < DD) {
#pragma unroll
    for (int r = 0; r < 8; ++r) {
      int mr = m0 + r + 8 * lhi;                        // D layout: vgpr r -> M=r / M=r+8
      ht[(size_t)mr * DD + c] = d[r];
    }
  }
}

// ---------------- fused attention pooling (gate GEMM in-register, 2-pass softmax) ----
__global__ __launch_bounds__(256) void att_pool_kernel(const float* __restrict__ xcur,
                                                       const float* __restrict__ maskf,
                                                       const float* __restrict__ Wg,
                                                       const float* __restrict__ bg,
                                                       float* __restrict__ out) {
  __shared__ float sh_h[NPG * DD];                      // 16 KB
  __shared__ float sh_m[NPG];
  __shared__ float sh_pmax[8][GATE];
  __shared__ float sh_gmax[GATE];
  __shared__ float sh_psum[8][GATE];
  __shared__ float sh_pwsum[8][GATE];

  int g = blockIdx.x, tid = threadIdx.x;
  int wave = tid >> 5, lane = tid & 31;
  int lhi = lane >> 4, khalf = lhi * 2;
  int nodeBase = g * NPG;

  const float4* src4 = (const float4*)(xcur + (size_t)nodeBase * DD);
  float4* dst4 = (float4*)sh_h;
  for (int i = tid; i < NPG * DD / 4; i += 256) dst4[i] = src4[i];
  for (int i = tid; i < NPG; i += 256) sh_m[i] = maskf[nodeBase + i];
  __syncthreads();

  // ---- pass 1: per-channel masked max ----
  for (int ct = 0; ct < 4; ++ct) {
    int c = ct * 16 + (lane & 15);
    float colb = bg[c];
    float cmax = NEGF;
#pragma unroll
    for (int mt = 0; mt < 4; ++mt) {
      int m = wave * 64 + mt * 16 + (lane & 15);
      v2f a0, a1, b0, b1;
      a0.x = sh_h[m * DD + khalf + 0]; a0.y = sh_h[m * DD + khalf + 1];
      a1.x = sh_h[m * DD + 4 + khalf + 0]; a1.y = sh_h[m * DD + 4 + khalf + 1];
      b0.x = Wg[(khalf + 0) * GATE + c]; b0.y = Wg[(khalf + 1) * GATE + c];
      b1.x = Wg[(4 + khalf + 0) * GATE + c]; b1.y = Wg[(4 + khalf + 1) * GATE + c];
      v8f d = {};
      d = wmma_f32_16x16x4(a0, b0, d);
      d = wmma_f32_16x16x4(a1, b1, d);
#pragma unroll
      for (int r = 0; r < 8; ++r) {
        int mr = wave * 64 + mt * 16 + r + 8 * lhi;
        float gv = (sh_m[mr] > 0.0f) ? (d[r] + colb) : NEGF;
        cmax = fmaxf(cmax, gv);
      }
    }
    cmax = fmaxf(cmax, __shfl_xor(cmax, 16, 32));       // lanes l / l+16 share col
    if (lane < 16) sh_pmax[wave][ct * 16 + lane] = cmax;
  }
  __syncthreads();
  if (tid < GATE) {
    float m = NEGF;
#pragma unroll
    for (int w = 0; w < 8; ++w) m = fmaxf(m, sh_pmax[w][tid]);
    sh_gmax[tid] = m;
  }
  __syncthreads();

  // ---- pass 2: recompute gate, accumulate e and e*gate ----
  for (int ct = 0; ct < 4; ++ct) {
    int c = ct * 16 + (lane & 15);
    float colb = bg[c];
    float gm = sh_gmax[c];
    float se = 0.0f, sw = 0.0f;
#pragma unroll
    for (int mt = 0; mt < 4; ++mt) {
      int m = wave * 64 + mt * 16 + (lane & 15);
      v2f a0, a1, b0, b1;
      a0.x = sh_h[m * DD + khalf + 0]; a0.y = sh_h[m * DD + khalf + 1];
      a1.x = sh_h[m * DD + 4 + khalf + 0]; a1.y = sh_h[m * DD + 4 + khalf + 1];
      b0.x = Wg[(khalf + 0) * GATE + c]; b0.y = Wg[(khalf + 1) * GATE + c];
      b1.x = Wg[(4 + khalf + 0) * GATE + c]; b1.y = Wg[(4 + khalf + 1) * GATE + c];
      v8f d = {};
      d = wmma_f32_16x16x4(a0, b0, d);
      d = wmma_f32_16x16x4(a1, b1, d);
#pragma unroll
      for (int r = 0; r < 8; ++r) {
        int mr = wave * 64 + mt * 16 + r + 8 * lhi;
        float gv = d[r] + colb;
        float e = (sh_m[mr] > 0.0f) ? __expf(gv - gm) : 0.0f;
        se += e;
        sw += e * gv;
      }
    }
    se += __shfl_xor(se, 16, 32);
    sw += __shfl_xor(sw, 16, 32);
    if (lane < 16) { sh_psum[wave][ct * 16 + lane] = se; sh_pwsum[wave][ct * 16 + lane] = sw; }
  }
  __syncthreads();
  if (tid < GATE) {
    float se = 0.0f, sw = 0.0f;
#pragma unroll
    for (int w = 0; w < 8; ++w) { se += sh_psum[w][tid]; sw += sh_pwsum[w][tid]; }
    out[g * GATE + tid] += sw / se;                     // accumulate across layers
  }
}

// ---------------- host-side launch ----------------
extern "C" void kernel_launch(void* const* d_in, const int* in_sizes, int n_in,
                              void* d_out, int out_size, void* d_ws, size_t ws_size,
                              hipStream_t stream) {
  const float* x    = (const float*)d_in[0];   // [N,8]
  const float* Wc   = (const float*)d_in[1];   // [4,8,8]
  const float* bc   = (const float*)d_in[2];   // [4,8]
  const float* pvec = (const float*)d_in[3];   // [4,8]
  const float* Wg   = (const float*)d_in[4];   // [8,64]
  const float* bg   = (const float*)d_in[5];   // [64]
  const int*   ei   = (const int*)d_in[6];     // [2,E]
  const int E = in_sizes[6] / 2;
  const int* row = ei;
  const int* col = ei + E;

  float* ws    = (float*)d_ws;
  float* xcur  = ws;                   // N*8
  float* ht    = xcur + (size_t)NN * DD;
  float* agg   = ht   + (size_t)NN * DD;
  float* deg   = agg  + (size_t)NN * DD;   // holds dinv after dinv_kernel
  float* score = deg  + NN;
  float* maskf = score + NN;
  float* out   = (float*)d_out;

  static const int KS[4] = {410, 328, 263, 211};

  const int T = 256;
  init_state_kernel<<<(NN * DD) / T, T, 0, stream>>>(x, xcur, maskf, out);

  for (int L = 0; L < 4; ++L) {
    const float* W = Wc + L * DD * DD;
    const float* b = bc + L * DD;
    const float* p = pvec + L * DD;

    deg_init_kernel<<<NN / T, T, 0, stream>>>(maskf, deg);
    edge_deg_kernel<<<(E + T - 1) / T, T, 0, stream>>>(row, col, maskf, deg, E);
    dinv_kernel<<<NN / T, T, 0, stream>>>(deg);

    gemm_xW_kernel<<<NN / 128, T, 0, stream>>>(xcur, W, ht);   // 16 nodes/wave, 8 waves/blk

    agg_init_kernel<<<(NN * DD) / T, T, 0, stream>>>(ht, deg, agg);
    edge_scatter_kernel<<<(E + T - 1) / T, T, 0, stream>>>(row, col, deg, ht, agg, E);
    bias_relu_kernel<<<(NN * DD) / T, T, 0, stream>>>(agg, b, maskf, xcur);

    score_kernel<<<NN / T, T, 0, stream>>>(xcur, p, score);
    topk_kernel<<<BB, NPG, 0, stream>>>(score, maskf, xcur, KS[L]);

    att_pool_kernel<<<BB, T, 0, stream>>>(xcur, maskf, Wg, bg, out);
  }
}